// SinkhornDistance_7138235646409
// MI455X (gfx1250) — compile-verified
//
#include <hip/hip_runtime.h>
#include <math.h>

// Sinkhorn distance, N=8192, D=256, EPS=0.1, 10 iterations.
// Strategy: never materialize the 256MB cost matrix. Recompute 16x16 dot tiles
// with V_WMMA_F32_16X16X4_F32 (full f32 precision) each pass; x,y (16MB total)
// stay L2-resident on MI455X (192MB L2), so all 21 passes are compute-fed from
// cache instead of 5.4GB of HBM re-reads.

#define NN 8192
#define DD 256
#define EPSV 0.1f
#define NCHUNK 8
#define COLS_PER_CHUNK (NN / NCHUNK) /* 1024 */
#define CTILES (COLS_PER_CHUNK / 16) /* 64 */
#define KSTEPS (DD / 4)              /* 64 */

typedef float v2f __attribute__((ext_vector_type(2)));
typedef float v8f __attribute__((ext_vector_type(8)));

__device__ __forceinline__ void lse_merge(float& m, float& s, float om, float os) {
  // Merge running logsumexp state (m,s) with (om,os). Requires at least one of
  // m/om finite (guaranteed by construction: first merge always brings a
  // finite value against the -inf/0 init).
  float nm = fmaxf(m, om);
  s = s * __expf(m - nm) + os * __expf(om - nm);
  m = nm;
}

// ---------------------------------------------------------------------------
// Row squared-norms: one 256-thread block per row (D == 256).
__global__ __launch_bounds__(256) void row_norms_kernel(const float* __restrict__ X,
                                                        float* __restrict__ out) {
  int row = blockIdx.x;
  float v = X[(size_t)row * DD + threadIdx.x];
  v *= v;
#pragma unroll
  for (int off = 16; off > 0; off >>= 1) v += __shfl_xor(v, off, 32);
  __shared__ float red[8];
  int wid = threadIdx.x >> 5;
  if ((threadIdx.x & 31) == 0) red[wid] = v;
  __syncthreads();
  if (threadIdx.x == 0) {
    float s = 0.0f;
#pragma unroll
    for (int i = 0; i < 8; ++i) s += red[i];
    out[row] = s;
  }
}

__global__ void zero_kernel(float* __restrict__ p, int n) {
  int i = blockIdx.x * blockDim.x + threadIdx.x;
  if (i < n) p[i] = 0.0f;
}

// ---------------------------------------------------------------------------
// Fused LSE pass:  partial(m,s) for  out[i] ~ logsumexp_j( v[j] - C_ij/EPS )
// with the row-constant -a2_i/EPS pulled out (applied in combine_kernel).
// t_ij = (v[j] - b2[j]/EPS) + (2/EPS) * dot(A_i, B_j).
// One wave (32 threads) per (16-row block, 1024-column chunk).
__global__ __launch_bounds__(32) void sinkhorn_lse_pass(
    const float* __restrict__ A, const float* __restrict__ B,
    const float* __restrict__ b2, const float* __restrict__ v,
    float* __restrict__ pm, float* __restrict__ ps) {
  const int lane = threadIdx.x;
  const int row_blk = blockIdx.x;   // 0..511
  const int chunk = blockIdx.y;     // 0..7
  const int row_base = row_blk * 16;
  const int col_begin = chunk * COLS_PER_CHUNK;
  const int lrow = lane & 15;
  const int hf = lane >> 4;
  const int khalf = hf * 2;  // ISA A/B layout: lanes 0-15 hold K={0,1}, 16-31 hold K={2,3}

  // Cache the A fragment (16 rows x 256 K) in registers: 128 VGPRs/lane,
  // reused across all 64 column tiles of this chunk.
  v2f afrag[KSTEPS];
  const float* arow = A + (size_t)(row_base + lrow) * DD;
#pragma unroll
  for (int t = 0; t < KSTEPS; ++t) afrag[t] = *(const v2f*)(arow + 4 * t + khalf);

  float m[8], s[8];
#pragma unroll
  for (int r = 0; r < 8; ++r) { m[r] = -INFINITY; s[r] = 0.0f; }

  for (int ct = 0; ct < CTILES; ++ct) {
    const int bn = col_begin + ct * 16 + lrow;
    const float* brow = B + (size_t)bn * DD;
    v8f acc = {};
#pragma unroll
    for (int t = 0; t < KSTEPS; ++t) {
      v2f bf = *(const v2f*)(brow + 4 * t + khalf);
      acc = __builtin_amdgcn_wmma_f32_16x16x4_f32(false, afrag[t], false, bf,
                                                  (short)0, acc, false, false);
    }
    const float w = v[bn] - b2[bn] * (1.0f / EPSV);
#pragma unroll
    for (int r = 0; r < 8; ++r) {
      float t_ = fmaf(acc[r], 2.0f / EPSV, w);
      lse_merge(m[r], s[r], t_, 1.0f);
    }
  }
  // Cross-lane merge across the 16 N-lanes of each half (halves track rows
  // r and r+8 respectively; xor masks 1..8 never cross the 16-lane boundary).
#pragma unroll
  for (int r = 0; r < 8; ++r) {
#pragma unroll
    for (int off = 1; off <= 8; off <<= 1) {
      float om = __shfl_xor(m[r], off, 32);
      float os = __shfl_xor(s[r], off, 32);
      lse_merge(m[r], s[r], om, os);
    }
  }
  if (lrow == 0) {
#pragma unroll
    for (int r = 0; r < 8; ++r) {
      int row = row_base + hf * 8 + r;
      pm[(size_t)row * NCHUNK + chunk] = m[r];
      ps[(size_t)row * NCHUNK + chunk] = s[r];
    }
  }
}

// Combine chunk partials:  out[i] = a2[i] - EPS*(M + log S).
__global__ __launch_bounds__(256) void combine_kernel(
    const float* __restrict__ pm, const float* __restrict__ ps,
    const float* __restrict__ a2, float* __restrict__ out) {
  int i = blockIdx.x * blockDim.x + threadIdx.x;
  if (i >= NN) return;
  float M = -INFINITY, S = 0.0f;
#pragma unroll
  for (int c = 0; c < NCHUNK; ++c)
    lse_merge(M, S, pm[(size_t)i * NCHUNK + c], ps[(size_t)i * NCHUNK + c]);
  out[i] = a2[i] - EPSV * (M + __logf(S));
}

// ---------------------------------------------------------------------------
// Final pass: global logsumexp over t_ij = (f_i + g_j - C_ij)/EPS + log(C_ij+1e-8),
// with C_ij = x2_i + y2_j - 2*dot recomputed via WMMA. One (m,s) per wave.
__global__ __launch_bounds__(32) void final_pass_kernel(
    const float* __restrict__ x, const float* __restrict__ y,
    const float* __restrict__ x2, const float* __restrict__ y2,
    const float* __restrict__ f, const float* __restrict__ g,
    float* __restrict__ qm, float* __restrict__ qs) {
  const int lane = threadIdx.x;
  const int row_blk = blockIdx.x;
  const int chunk = blockIdx.y;
  const int row_base = row_blk * 16;
  const int col_begin = chunk * COLS_PER_CHUNK;
  const int lrow = lane & 15;
  const int hf = lane >> 4;
  const int khalf = hf * 2;

  v2f afrag[KSTEPS];
  const float* arow = x + (size_t)(row_base + lrow) * DD;
#pragma unroll
  for (int t = 0; t < KSTEPS; ++t) afrag[t] = *(const v2f*)(arow + 4 * t + khalf);

  float fi[8], xi2[8];
#pragma unroll
  for (int r = 0; r < 8; ++r) {
    int row = row_base + hf * 8 + r;  // C/D layout: VGPR r -> row r (+8 for upper half)
    fi[r] = f[row];
    xi2[r] = x2[row];
  }

  float m[8], s[8];
#pragma unroll
  for (int r = 0; r < 8; ++r) { m[r] = -INFINITY; s[r] = 0.0f; }

  for (int ct = 0; ct < CTILES; ++ct) {
    const int bn = col_begin + ct * 16 + lrow;
    const float* brow = y + (size_t)bn * DD;
    v8f acc = {};
#pragma unroll
    for (int t = 0; t < KSTEPS; ++t) {
      v2f bf = *(const v2f*)(brow + 4 * t + khalf);
      acc = __builtin_amdgcn_wmma_f32_16x16x4_f32(false, afrag[t], false, bf,
                                                  (short)0, acc, false, false);
    }
    const float gj = g[bn];
    const float yj2 = y2[bn];
#pragma unroll
    for (int r = 0; r < 8; ++r) {
      float C = xi2[r] + yj2 - 2.0f * acc[r];
      float t_ = (fi[r] + gj - C) * (1.0f / EPSV) + __logf(fmaxf(C, 0.0f) + 1e-8f);
      lse_merge(m[r], s[r], t_, 1.0f);
    }
  }
  // Fold the 8 row-slots, then all 32 lanes (global LSE, rows may mix freely).
  float M = m[0], S = s[0];
#pragma unroll
  for (int r = 1; r < 8; ++r) lse_merge(M, S, m[r], s[r]);
#pragma unroll
  for (int off = 1; off < 32; off <<= 1) {
    float om = __shfl_xor(M, off, 32);
    float os = __shfl_xor(S, off, 32);
    lse_merge(M, S, om, os);
  }
  if (lane == 0) {
    int idx = row_blk * NCHUNK + chunk;
    qm[idx] = M;
    qs[idx] = S;
  }
}

// Fallback: per-block partial sums of (x-y)^2 (deterministic, no float atomics).
__global__ __launch_bounds__(256) void sqdiff_partial_kernel(
    const float* __restrict__ x, const float* __restrict__ y, float* __restrict__ part) {
  int i = blockIdx.x * blockDim.x + threadIdx.x;
  float d = x[i] - y[i];
  float v = d * d;
#pragma unroll
  for (int off = 16; off > 0; off >>= 1) v += __shfl_xor(v, off, 32);
  __shared__ float red[8];
  int wid = threadIdx.x >> 5;
  if ((threadIdx.x & 31) == 0) red[wid] = v;
  __syncthreads();
  if (threadIdx.x == 0) {
    float sum = 0.0f;
#pragma unroll
    for (int k = 0; k < 8; ++k) sum += red[k];
    part[blockIdx.x] = sum;
  }
}

__global__ __launch_bounds__(256) void finalize_kernel(
    const float* __restrict__ qm, const float* __restrict__ qs,
    const float* __restrict__ sqp, float* __restrict__ out) {
  int t = threadIdx.x;
  float M = -INFINITY, S = 0.0f;
  for (int i = t; i < NN / 16 * NCHUNK; i += 256) lse_merge(M, S, qm[i], qs[i]);
  float q = 0.0f;
  for (int i = t; i < NN; i += 256) q += sqp[i];
  __shared__ float sm[256], ss[256], sq[256];
  sm[t] = M; ss[t] = S; sq[t] = q;
  __syncthreads();
  for (int stride = 128; stride > 0; stride >>= 1) {
    if (t < stride) {
      float m0 = sm[t], s0 = ss[t];
      lse_merge(m0, s0, sm[t + stride], ss[t + stride]);
      sm[t] = m0; ss[t] = s0;
      sq[t] += sq[t + stride];
    }
    __syncthreads();
  }
  if (t == 0) {
    float lse = sm[0] + __logf(ss[0]);
    float log_dist = lse - 2.0f * __logf((float)NN);
    float dist = __expf(log_dist);
    float fallback = sq[0] / (float)(NN * DD);
    bool bad = !isfinite(dist) || (dist > 1e6f);
    out[0] = bad ? fallback : dist;
  }
}

// ---------------------------------------------------------------------------
extern "C" void kernel_launch(void* const* d_in, const int* in_sizes, int n_in,
                              void* d_out, int out_size, void* d_ws, size_t ws_size,
                              hipStream_t stream) {
  (void)in_sizes; (void)n_in; (void)out_size; (void)ws_size;
  const float* x = (const float*)d_in[0];
  const float* y = (const float*)d_in[1];
  float* ws = (float*)d_ws;

  // Workspace layout (floats): ~720KB total.
  float* f  = ws;                       // NN
  float* g  = f + NN;                   // NN
  float* x2 = g + NN;                   // NN
  float* y2 = x2 + NN;                  // NN
  float* pm = y2 + NN;                  // NN*NCHUNK
  float* ps = pm + (size_t)NN * NCHUNK; // NN*NCHUNK
  float* qm = ps + (size_t)NN * NCHUNK; // (NN/16)*NCHUNK = 4096
  float* qs = qm + (NN / 16) * NCHUNK;  // 4096
  float* sqp = qs + (NN / 16) * NCHUNK; // NN partial sums for fallback

  // Init f,g = 0 (contiguous 2*NN floats).
  zero_kernel<<<(2 * NN + 255) / 256, 256, 0, stream>>>(f, 2 * NN);

  // Row squared-norms.
  row_norms_kernel<<<NN, 256, 0, stream>>>(x, x2);
  row_norms_kernel<<<NN, 256, 0, stream>>>(y, y2);

  dim3 passGrid(NN / 16, NCHUNK);  // (512, 8) waves
  for (int it = 0; it < 10; ++it) {
    // f_i = x2_i - EPS * LSE_j( (g_j - y2_j/EPS) + (2/EPS) x_i.y_j )
    sinkhorn_lse_pass<<<passGrid, 32, 0, stream>>>(x, y, y2, g, pm, ps);
    combine_kernel<<<NN / 256, 256, 0, stream>>>(pm, ps, x2, f);
    // g_j = y2_j - EPS * LSE_i( (f_i - x2_i/EPS) + (2/EPS) y_j.x_i )   (uses new f)
    sinkhorn_lse_pass<<<passGrid, 32, 0, stream>>>(y, x, x2, f, pm, ps);
    combine_kernel<<<NN / 256, 256, 0, stream>>>(pm, ps, y2, g);
  }

  final_pass_kernel<<<passGrid, 32, 0, stream>>>(x, y, x2, y2, f, g, qm, qs);
  sqdiff_partial_kernel<<<NN, 256, 0, stream>>>(x, y, sqp);
  finalize_kernel<<<1, 256, 0, stream>>>(qm, qs, sqp, (float*)d_out);
}